// EdgeToGlobal_14620068675882
// MI455X (gfx1250) — compile-verified
//
#include <hip/hip_runtime.h>
#include <math.h>

typedef __attribute__((ext_vector_type(2))) float v2f;
typedef __attribute__((ext_vector_type(8))) float v8f;

#define EDIM 128
#define GDIM 256
#define BNUM 1024
#define RTHREADS 256   // 8 waves per graph-block
#define RTILE 256      // edge ids staged per LDS tile

// ---------------------------------------------------------------------------
// Pass 1: per-edge graph id + histogram (4M int atomics total, not 1B float)
// ---------------------------------------------------------------------------
__global__ void k_ebatch(const long long* __restrict__ e_index,
                         const long long* __restrict__ batch,
                         int* __restrict__ e_batch,
                         int* __restrict__ count, int E) {
  int i = blockIdx.x * blockDim.x + threadIdx.x;
  if (i >= E) return;
  long long node = e_index[(long long)E + i];   // row 1 of e_index
  int eb = (int)batch[node];
  e_batch[i] = eb;
  atomicAdd(&count[eb], 1);
}

// ---------------------------------------------------------------------------
// Pass 2: exclusive scan over B=1024 counts (single workgroup, LDS scan)
// ---------------------------------------------------------------------------
__global__ void k_scan(const int* __restrict__ count,
                       int* __restrict__ offs,
                       int* __restrict__ cursor) {
  __shared__ int tmp[BNUM];
  int t = threadIdx.x;
  int c = count[t];
  tmp[t] = c;
  __syncthreads();
  for (int d = 1; d < BNUM; d <<= 1) {
    int v = (t >= d) ? tmp[t - d] : 0;
    __syncthreads();
    tmp[t] += v;
    __syncthreads();
  }
  int excl = tmp[t] - c;
  offs[t] = excl;
  cursor[t] = excl;
}

// ---------------------------------------------------------------------------
// Pass 3: scatter edge ids into graph-sorted order
// ---------------------------------------------------------------------------
__global__ void k_scatter(const int* __restrict__ e_batch,
                          int* __restrict__ cursor,
                          int* __restrict__ sorted, int E) {
  int i = blockIdx.x * blockDim.x + threadIdx.x;
  if (i >= E) return;
  int eb = e_batch[i];
  int pos = atomicAdd(&cursor[eb], 1);
  sorted[pos] = i;
}

// ---------------------------------------------------------------------------
// Pass 4: one 256-thread block per graph. 32 lanes x float4 = 128 dims per
// edge (global_load_b128, 512B per wave-issue); the block's 8 waves
// interleave over edges. Single streaming pass over e (1 GB): sum, sumsq,
// min, max in float4 registers; std via sumsq - c*mean^2 (no second pass).
// Cross-wave combine through an 8x128 LDS tree at the end.
// ---------------------------------------------------------------------------
__global__ void k_reduce(const float* __restrict__ e,
                         const int* __restrict__ sorted,
                         const int* __restrict__ offs,
                         const int* __restrict__ count,
                         float* __restrict__ g) {
  int b    = blockIdx.x;
  int t    = threadIdx.x;        // 0..255
  int grp  = t >> 5;             // 0..7: edge-interleave group (== wave id)
  int lane = t & 31;             // dim quad: dims [4*lane, 4*lane+4)
  int start = offs[b];
  int cnt   = count[b];

  float4 sum = {0.f, 0.f, 0.f, 0.f};
  float4 sq  = {0.f, 0.f, 0.f, 0.f};
  float4 mn  = { INFINITY,  INFINITY,  INFINITY,  INFINITY};
  float4 mx  = {-INFINITY, -INFINITY, -INFINITY, -INFINITY};

  __shared__ int eids[RTILE];

  for (int base = 0; base < cnt; base += RTILE) {
    int m = min(RTILE, cnt - base);
    __syncthreads();
    if (t < m) eids[t] = sorted[start + base + t];
    __syncthreads();
    for (int j = grp; j < m; j += 8) {          // j uniform per wave
      const float4* row = (const float4*)(e + (size_t)eids[j] * EDIM);
      if (j + 32 < m) {                          // prefetch 4 iterations ahead
        __builtin_prefetch((const float4*)(e + (size_t)eids[j + 32] * EDIM) + lane, 0, 1);
      }
      float4 x = row[lane];
      sum.x += x.x;  sum.y += x.y;  sum.z += x.z;  sum.w += x.w;
      sq.x = fmaf(x.x, x.x, sq.x);  sq.y = fmaf(x.y, x.y, sq.y);
      sq.z = fmaf(x.z, x.z, sq.z);  sq.w = fmaf(x.w, x.w, sq.w);
      mn.x = fminf(mn.x, x.x);  mn.y = fminf(mn.y, x.y);
      mn.z = fminf(mn.z, x.z);  mn.w = fminf(mn.w, x.w);
      mx.x = fmaxf(mx.x, x.x);  mx.y = fmaxf(mx.y, x.y);
      mx.z = fmaxf(mx.z, x.z);  mx.w = fmaxf(mx.w, x.w);
    }
  }

  // --- cross-group combine: 8 partials per dim, one quantity at a time ---
  __shared__ float red[8][EDIM];
  int d0 = lane * 4;
  float S = 0.f, Q = 0.f, MN = 0.f, MX = 0.f;

  __syncthreads();
  red[grp][d0] = sum.x; red[grp][d0+1] = sum.y; red[grp][d0+2] = sum.z; red[grp][d0+3] = sum.w;
  __syncthreads();
  if (t < EDIM) { S = red[0][t]; for (int k = 1; k < 8; ++k) S += red[k][t]; }
  __syncthreads();

  red[grp][d0] = sq.x; red[grp][d0+1] = sq.y; red[grp][d0+2] = sq.z; red[grp][d0+3] = sq.w;
  __syncthreads();
  if (t < EDIM) { Q = red[0][t]; for (int k = 1; k < 8; ++k) Q += red[k][t]; }
  __syncthreads();

  red[grp][d0] = mn.x; red[grp][d0+1] = mn.y; red[grp][d0+2] = mn.z; red[grp][d0+3] = mn.w;
  __syncthreads();
  if (t < EDIM) { MN = red[0][t]; for (int k = 1; k < 8; ++k) MN = fminf(MN, red[k][t]); }
  __syncthreads();

  red[grp][d0] = mx.x; red[grp][d0+1] = mx.y; red[grp][d0+2] = mx.z; red[grp][d0+3] = mx.w;
  __syncthreads();
  if (t < EDIM) { MX = red[0][t]; for (int k = 1; k < 8; ++k) MX = fmaxf(MX, red[k][t]); }

  if (t < EDIM) {
    float c = (float)cnt;
    float mean = S / fmaxf(c, 1.0f);
    float var_sum = fmaxf(Q - c * mean * mean, 0.0f);
    float denom = fmaxf(c - 1.0f, 1.0f) + 1e-6f;
    float sd = sqrtf(var_sum / denom);
    if (cnt == 0) { MN = 0.0f; MX = 0.0f; }   // torch_scatter empty-segment fill

    size_t row = (size_t)b * (4 * EDIM);
    g[row + t]            = mean;
    g[row + EDIM + t]     = MN;
    g[row + 2 * EDIM + t] = MX;
    g[row + 3 * EDIM + t] = sd;
  }
}

// ---------------------------------------------------------------------------
// Pass 5: out = g[1024x512] @ fc_w[512x256] + fc_b, via V_WMMA_F32_16X16X4_F32
// One wave (32 threads) per 16x16 output tile; 128 K=4 WMMA steps.
// Fragment layouts per CDNA5 ISA 7.12.2 (f32 A 16x4 = v2f, B 4x16 = v2f,
// C/D 16x16 = v8f with M=r (lanes 0-15) / M=r+8 (lanes 16-31)).
// ---------------------------------------------------------------------------
__global__ void k_gemm_wmma(const float* __restrict__ g,
                            const float* __restrict__ w,
                            const float* __restrict__ bias,
                            float* __restrict__ out) {
  int tileN = blockIdx.x;        // 0..15  (256 cols / 16)
  int tileM = blockIdx.y;        // 0..63  (1024 rows / 16)
  int lane  = threadIdx.x;       // 0..31, wave32
  int half  = lane >> 4;         // K sub-split
  int l     = lane & 15;

  int rowA = tileM * 16 + l;     // A row this lane owns
  int col  = tileN * 16 + l;     // B/C/D column this lane owns

  const float* arow = g + (size_t)rowA * (4 * EDIM);
  v8f acc = {};

  for (int k0 = 0; k0 < 4 * EDIM; k0 += 4) {
    int ka = k0 + 2 * half;
    v2f a, bf;
    a.x  = arow[ka];                        // A[rowA][ka]
    a.y  = arow[ka + 1];                    // A[rowA][ka+1]
    bf.x = w[(size_t)ka * GDIM + col];      // B[ka][col]
    bf.y = w[(size_t)(ka + 1) * GDIM + col];
    acc = __builtin_amdgcn_wmma_f32_16x16x4_f32(
        /*neg_a=*/false, a, /*neg_b=*/false, bf,
        /*c_mod=*/(short)0, acc, /*reuse_a=*/false, /*reuse_b=*/false);
  }

  float bb = bias[col];
#pragma unroll
  for (int r = 0; r < 8; ++r) {
    int row = tileM * 16 + r + 8 * half;    // C/D layout: M=r / M=r+8
    out[(size_t)row * GDIM + col] = acc[r] + bb;
  }
}

// ---------------------------------------------------------------------------
extern "C" void kernel_launch(void* const* d_in, const int* in_sizes, int n_in,
                              void* d_out, int out_size, void* d_ws, size_t ws_size,
                              hipStream_t stream) {
  const long long* e_index = (const long long*)d_in[0];  // [2,E] int64
  const float*     e       = (const float*)d_in[1];      // [E,128] f32
  const long long* batch   = (const long long*)d_in[2];  // [N] int64
  const float*     fc_w    = (const float*)d_in[3];      // [512,256] f32
  const float*     fc_b    = (const float*)d_in[4];      // [256] f32
  float*           out     = (float*)d_out;              // [1024,256] f32

  int E = in_sizes[0] / 2;

  // Workspace layout (bytes)
  char* ws = (char*)d_ws;
  size_t eAligned = ((size_t)E * sizeof(int) + 255) & ~(size_t)255;
  int*   count   = (int*)(ws + 0);
  int*   offs    = (int*)(ws + 4096);
  int*   cursor  = (int*)(ws + 8192);
  int*   e_batch = (int*)(ws + 65536);
  int*   sorted  = (int*)(ws + 65536 + eAligned);
  float* g       = (float*)(ws + 65536 + 2 * eAligned);  // [1024,512]

  hipMemsetAsync(count, 0, BNUM * sizeof(int), stream);

  int threads = 256;
  int eblocks = (E + threads - 1) / threads;
  k_ebatch <<<eblocks, threads, 0, stream>>>(e_index, batch, e_batch, count, E);
  k_scan   <<<1, BNUM, 0, stream>>>(count, offs, cursor);
  k_scatter<<<eblocks, threads, 0, stream>>>(e_batch, cursor, sorted, E);
  k_reduce <<<BNUM, RTHREADS, 0, stream>>>(e, sorted, offs, count, g);

  dim3 ggrid(GDIM / 16, BNUM / 16);  // (16, 64) tiles
  k_gemm_wmma<<<ggrid, 32, 0, stream>>>(g, fc_w, fc_b, out);
}